// GCN_60232621359253
// MI455X (gfx1250) — compile-verified
//
#include <hip/hip_runtime.h>
#include <hip/hip_bf16.h>

typedef __attribute__((ext_vector_type(2))) float v2f;
typedef __attribute__((ext_vector_type(8))) float v8f;

#define N_NODES 100000
#define N_EDGES 1600000

// ---------------------------------------------------------------------------
// Zero a float buffer, float4 per thread (all sizes here are multiples of 4)
// ---------------------------------------------------------------------------
__global__ void gcn_zero4(float4* __restrict__ p, size_t nquads) {
    size_t i = (size_t)blockIdx.x * blockDim.x + threadIdx.x;
    if (i < nquads) p[i] = make_float4(0.f, 0.f, 0.f, 0.f);
}

// ---------------------------------------------------------------------------
// In-degree accumulation: deg[dst[e]] += 1
// ---------------------------------------------------------------------------
__global__ void gcn_degree(const int* __restrict__ dst, float* __restrict__ deg, int e_count) {
    int e = blockIdx.x * blockDim.x + threadIdx.x;
    if (e < e_count) unsafeAtomicAdd(&deg[dst[e]], 1.0f);
}

// ---------------------------------------------------------------------------
// dinv = deg > 0 ? rsqrt(deg) : 0
// ---------------------------------------------------------------------------
__global__ void gcn_dinv(const float* __restrict__ deg, float* __restrict__ dinv, int n) {
    int i = blockIdx.x * blockDim.x + threadIdx.x;
    if (i < n) {
        float d = deg[i];
        dinv[i] = (d > 0.0f) ? rsqrtf(d) : 0.0f;
    }
}

// ---------------------------------------------------------------------------
// Per-edge weight: wedge[e] = dinv[src[e]] * dinv[dst[e]]
// ---------------------------------------------------------------------------
__global__ void gcn_edge_w(const int* __restrict__ src, const int* __restrict__ dst,
                           const float* __restrict__ dinv, float* __restrict__ wedge,
                           int e_count) {
    int e = blockIdx.x * blockDim.x + threadIdx.x;
    if (e < e_count) wedge[e] = dinv[src[e]] * dinv[dst[e]];
}

// ---------------------------------------------------------------------------
// H[nRows, NOUT] = X[nRows, 128] @ W[128, NOUT] via V_WMMA_F32_16X16X4_F32.
// One wave per 16x32 output strip: two 16x16 accumulator tiles share each A
// fragment (2 WMMAs per A load). NOUT is a compile-time constant so all loads
// and stores use immediate IOFFSETs (no per-iteration 64-bit address math).
// VGPR layouts per CDNA5 ISA §7.12.2 (f32 tables):
//   A 16x4 : lane l -> row (l&15); vgpr0: K = 2*(l>>4), vgpr1: K = 2*(l>>4)+1
//   B 4x16 : lane l -> col (l&15); vgpr0: K = 2*(l>>4), vgpr1: K = 2*(l>>4)+1
//   C 16x16: lane l, vgpr r -> row r + 8*(l>>4), col (l&15)
// blockDim == 32, no divergence -> EXEC all ones as WMMA requires.
// ---------------------------------------------------------------------------
template <int NOUT>
__global__ void gcn_gemm_wmma(const float* __restrict__ X, const float* __restrict__ W,
                              float* __restrict__ H) {
    const int lane = threadIdx.x & 31;
    const int half = lane >> 4;   // 0 or 1
    const int l15  = lane & 15;
    const int m0 = blockIdx.x * 16;
    const int n0 = blockIdx.y * 32;

    const float* __restrict__ arow = X + (size_t)(m0 + l15) * 128 + half * 2;
    const float* __restrict__ bcol = W + (size_t)(half * 2) * NOUT + n0 + l15;

    v8f c0 = {};
    v8f c1 = {};
#pragma unroll
    for (int k = 0; k < 128; k += 4) {
        v2f a, b0, b1;
        a.x  = arow[k + 0];               // global_load_b64, imm offset
        a.y  = arow[k + 1];
        b0.x = bcol[k * NOUT + 0];        // imm offsets (max 127*128*4 < 2^23)
        b0.y = bcol[k * NOUT + NOUT];
        b1.x = bcol[k * NOUT + 16];
        b1.y = bcol[k * NOUT + NOUT + 16];
        c0 = __builtin_amdgcn_wmma_f32_16x16x4_f32(false, a, false, b0, (short)0, c0, false, false);
        c1 = __builtin_amdgcn_wmma_f32_16x16x4_f32(false, a, false, b1, (short)0, c1, false, false);
    }

    float* __restrict__ hrow = H + (size_t)(m0 + half * 8) * NOUT + n0 + l15;
#pragma unroll
    for (int r = 0; r < 8; ++r) {
        hrow[r * NOUT + 0]  = c0[r];
        hrow[r * NOUT + 16] = c1[r];
    }
}

// ---------------------------------------------------------------------------
// Edge scatter: agg[dst[e], f] += h[src[e], f] * wedge[e]
// 128-thread block; each thread owns one float4 quad of one edge:
// global_load_b128 gather + 4x global_atomic_add_f32.
// ---------------------------------------------------------------------------
template <int F>
__global__ void gcn_scatter(const float* __restrict__ h, const int* __restrict__ src,
                            const int* __restrict__ dst, const float* __restrict__ wedge,
                            float* __restrict__ agg) {
    constexpr int QPE = F / 4;        // quads per edge (32 or 16)
    constexpr int EPB = 128 / QPE;    // edges per block (4 or 8)
    const int t  = threadIdx.x;
    const int le = t / QPE;
    const int q  = t % QPE;
    const size_t e = (size_t)blockIdx.x * EPB + le;
    const int s = src[e];
    const int d = dst[e];
    const float wv = wedge[e];
    const float4 hv = *(const float4*)(h + (size_t)s * F + q * 4);
    float* ap = agg + (size_t)d * F + q * 4;
    unsafeAtomicAdd(ap + 0, hv.x * wv);
    unsafeAtomicAdd(ap + 1, hv.y * wv);
    unsafeAtomicAdd(ap + 2, hv.z * wv);
    unsafeAtomicAdd(ap + 3, hv.w * wv);
}

// ---------------------------------------------------------------------------
// In-place bias (+ optional ReLU), float4 per thread.
// ---------------------------------------------------------------------------
template <int F, bool RELU>
__global__ void gcn_bias_act(float* __restrict__ buf, const float* __restrict__ b,
                             size_t nquads) {
    size_t i = (size_t)blockIdx.x * blockDim.x + threadIdx.x;
    if (i >= nquads) return;
    const int q = (int)(i % (F / 4));
    float4 v = *(float4*)(buf + i * 4);
    const float4 bb = *(const float4*)(b + q * 4);
    v.x += bb.x; v.y += bb.y; v.z += bb.z; v.w += bb.w;
    if (RELU) {
        v.x = fmaxf(v.x, 0.f); v.y = fmaxf(v.y, 0.f);
        v.z = fmaxf(v.z, 0.f); v.w = fmaxf(v.w, 0.f);
    }
    *(float4*)(buf + i * 4) = v;
}

// ---------------------------------------------------------------------------
extern "C" void kernel_launch(void* const* d_in, const int* in_sizes, int n_in,
                              void* d_out, int out_size, void* d_ws, size_t ws_size,
                              hipStream_t stream) {
    const float* x  = (const float*)d_in[0];
    const int*   ei = (const int*)d_in[1];
    const float* W1 = (const float*)d_in[2];
    const float* b1 = (const float*)d_in[3];
    const float* W2 = (const float*)d_in[4];
    const float* b2 = (const float*)d_in[5];
    const float* W3 = (const float*)d_in[6];
    const float* b3 = (const float*)d_in[7];
    float* out = (float*)d_out;

    const int N = N_NODES;
    const int E = N_EDGES;
    const int* src = ei;
    const int* dst = ei + E;

    // Workspace (floats): deg[N] | dinv[N] | wedge[E] | buf0[N*128] | buf1[N*128]
    float* ws    = (float*)d_ws;
    float* deg   = ws;
    float* dinv  = deg + N;
    float* wedge = dinv + N;
    float* buf0  = wedge + E;
    float* buf1  = buf0 + (size_t)N * 128;

    const size_t bigQ = (size_t)N * 128 / 4;   // quads in a 128-wide buffer
    const size_t outQ = (size_t)N * 64 / 4;
    const int ZB = 256;
    auto blocks = [](size_t n, int b) { return (unsigned)((n + b - 1) / b); };

    // --- Edge-normalization weights (computed once, reused by all 3 layers) ---
    gcn_zero4<<<blocks(N / 4, ZB), ZB, 0, stream>>>((float4*)deg, (size_t)N / 4);
    gcn_degree<<<blocks(E, ZB), ZB, 0, stream>>>(dst, deg, E);
    gcn_dinv<<<blocks(N, ZB), ZB, 0, stream>>>(deg, dinv, N);
    gcn_edge_w<<<blocks(E, ZB), ZB, 0, stream>>>(src, dst, dinv, wedge, E);

    dim3 g128(N / 16, 128 / 32);
    dim3 g64(N / 16, 64 / 32);

    // --- Layer 1: buf1 = relu(Agg(x @ W1) + b1) ---
    gcn_gemm_wmma<128><<<g128, 32, 0, stream>>>(x, W1, buf0);
    gcn_zero4<<<blocks(bigQ, ZB), ZB, 0, stream>>>((float4*)buf1, bigQ);
    gcn_scatter<128><<<(unsigned)(E / 4), 128, 0, stream>>>(buf0, src, dst, wedge, buf1);
    gcn_bias_act<128, true><<<blocks(bigQ, ZB), ZB, 0, stream>>>(buf1, b1, bigQ);

    // --- Layer 2: buf1 = relu(Agg(buf1 @ W2) + b2) (buf1 dead after GEMM) ---
    gcn_gemm_wmma<128><<<g128, 32, 0, stream>>>(buf1, W2, buf0);
    gcn_zero4<<<blocks(bigQ, ZB), ZB, 0, stream>>>((float4*)buf1, bigQ);
    gcn_scatter<128><<<(unsigned)(E / 4), 128, 0, stream>>>(buf0, src, dst, wedge, buf1);
    gcn_bias_act<128, true><<<blocks(bigQ, ZB), ZB, 0, stream>>>(buf1, b2, bigQ);

    // --- Layer 3: out = Agg(buf1 @ W3) + b3 (no relu) ---
    gcn_gemm_wmma<64><<<g64, 32, 0, stream>>>(buf1, W3, buf0);
    gcn_zero4<<<blocks(outQ, ZB), ZB, 0, stream>>>((float4*)out, outQ);
    gcn_scatter<64><<<(unsigned)(E / 8), 128, 0, stream>>>(buf0, src, dst, wedge, out);
    gcn_bias_act<64, false><<<blocks(outQ, ZB), ZB, 0, stream>>>(out, b3, outQ);
}